// ROLANDGNN_42073499631742
// MI455X (gfx1250) — compile-verified
//
#include <hip/hip_runtime.h>
#include <math.h>

// ---------------------------------------------------------------------------
// ROLAND GNN forward for MI455X (gfx1250, wave32).
// GEMMs use V_WMMA_F32_16X16X4_F32 (fp32 in/out, exact precision path), with
// each wave computing a 16 x (16*NT) output strip so one A fragment feeds NT
// WMMAs (A traffic / NT). GCN aggregation uses global_atomic_add_f32 scatter.
// ---------------------------------------------------------------------------

typedef float v2f __attribute__((ext_vector_type(2)));
typedef float v8f __attribute__((ext_vector_type(8)));

#define LSLOPE 0.01f

__device__ __forceinline__ float leaky_f(float v) { return v > 0.f ? v : LSLOPE * v; }

// ---------------------------------------------------------------------------
// WMMA GEMM: out[M,Ncols] = act(A[M,K] @ W[K,Ncols] + bias)
// One wave computes NT 16x16 output tiles (one 16 x 16*NT strip), sharing the
// A fragment. Requirements: M % 16 == 0, (Ncols/16) % NT == 0, K % 4 == 0.
//
// f32 WMMA fragment layout (ISA 7.12.2):
//   A 16x4 (2 VGPRs): lanes 0-15 -> M=lane, VGPR j holds K=k0+j; lanes 16-31
//                     hold K=k0+2+j for M=lane-16.
//   B 4x16 (2 VGPRs): mirrored, lane -> N column, lane-half selects K pair.
//   C/D 16x16 (8 VGPRs): VGPR r -> row m0 + r + 8*(lane>=16), col = n0+(lane&15)
// ---------------------------------------------------------------------------
template<int ACT, int HAS_BIAS, int NT>
__global__ __launch_bounds__(256)
void wmma_gemm_f32_kernel(const float* __restrict__ A,
                          const float* __restrict__ W,
                          const float* __restrict__ bias,
                          float* __restrict__ out,
                          int M, int K, int Ncols)
{
    const int lane    = threadIdx.x & 31;
    const int gwave   = blockIdx.x * (blockDim.x >> 5) + (threadIdx.x >> 5);
    const int ngrps   = (Ncols >> 4) / NT;
    const int totGrps = (M >> 4) * ngrps;
    if (gwave >= totGrps) return;             // wave-uniform: EXEC stays all-ones

    const int mtile = gwave / ngrps;
    const int ngrp  = gwave - mtile * ngrps;
    const int m0 = mtile << 4;
    const int n0 = (ngrp * NT) << 4;
    const int lm = lane & 15;                 // row (A) / col (B,C) within tile
    const int lh = lane >> 4;                 // lane-half selects K pair

    const float* __restrict__ arow = A + (size_t)(m0 + lm) * K;
    const float* __restrict__ wcol = W + n0 + lm;

    v8f acc[NT];
#pragma unroll
    for (int t = 0; t < NT; ++t) acc[t] = (v8f){};

#pragma unroll 2
    for (int k0 = 0; k0 < K; k0 += 4) {
        const int ka = k0 + (lh << 1);
        v2f a;
        a.x = arow[ka];
        a.y = arow[ka + 1];
        if (k0 + 16 < K) __builtin_prefetch(arow + ka + 16, 0, 1); // global_prefetch_b8
#pragma unroll
        for (int t = 0; t < NT; ++t) {
            v2f b;
            b.x = wcol[(size_t)ka * Ncols + 16 * t];
            b.y = wcol[(size_t)(ka + 1) * Ncols + 16 * t];
            // D = A * B + C  (fp32, 16x16x4); one A fragment feeds NT WMMAs
            acc[t] = __builtin_amdgcn_wmma_f32_16x16x4_f32(
                /*neg_a=*/false, a, /*neg_b=*/false, b,
                /*c_mod=*/(short)0, acc[t], /*reuse_a=*/false, /*reuse_b=*/false);
        }
    }

#pragma unroll
    for (int t = 0; t < NT; ++t) {
        const int nc = n0 + 16 * t + lm;
        const float bv = HAS_BIAS ? bias[nc] : 0.f;
        float* __restrict__ ocol = out + (size_t)nc;
#pragma unroll
        for (int r = 0; r < 8; ++r) {
            const int m = m0 + r + (lh << 3);
            float v = acc[t][r] + bv;
            if (ACT) v = leaky_f(v);
            ocol[(size_t)m * Ncols] = v;
        }
    }
}

// ---------------------------------------------------------------------------
// Elementwise / graph kernels
// ---------------------------------------------------------------------------
__global__ void fill_kernel(float* __restrict__ p, float v, int n)
{
    int i = blockIdx.x * blockDim.x + threadIdx.x;
    if (i < n) p[i] = v;
}

__global__ void deg_kernel(const int* __restrict__ dst, float* __restrict__ deg, int E)
{
    int e = blockIdx.x * blockDim.x + threadIdx.x;
    if (e < E) atomicAdd(&deg[dst[e]], 1.0f);
}

__global__ void dinv_kernel(const float* __restrict__ deg, float* __restrict__ dinv, int n)
{
    int i = blockIdx.x * blockDim.x + threadIdx.x;
    if (i < n) dinv[i] = rsqrtf(deg[i]);
}

// agg[dst] += hw[src] * dinv[src]*dinv[dst]; one thread per (edge, 4 feats)
template<int F>
__global__ void scatter_kernel(const float* __restrict__ hw,
                               const int* __restrict__ src,
                               const int* __restrict__ dst,
                               const float* __restrict__ dinv,
                               float* __restrict__ agg, int E)
{
    const int groups = F >> 2;
    int t = blockIdx.x * blockDim.x + threadIdx.x;
    int e = t / groups;
    if (e >= E) return;
    int q = (t - e * groups) << 2;
    int s = src[e], d = dst[e];
    float c = dinv[s] * dinv[d];
    const float4 hv = *(const float4*)(hw + (size_t)s * F + q);
    float* ap = agg + (size_t)d * F + q;
    atomicAdd(ap + 0, hv.x * c);
    atomicAdd(ap + 1, hv.y * c);
    atomicAdd(ap + 2, hv.z * c);
    atomicAdd(ap + 3, hv.w * c);
}

// agg = leaky(agg + hw*dinv^2 + bias)   (in-place over agg)
__global__ void gcn_finalize_kernel(float* __restrict__ agg,
                                    const float* __restrict__ hw,
                                    const float* __restrict__ dinv,
                                    const float* __restrict__ bias,
                                    int Nn, int F)
{
    int t = blockIdx.x * blockDim.x + threadIdx.x;
    if (t >= Nn * F) return;
    int i = t / F;
    int j = t - i * F;
    float di = dinv[i];
    float v = agg[t] + hw[t] * di * di + bias[j];
    agg[t] = leaky_f(v);
}

// torch GRUCell gates, (r,z,n) order
__global__ void gru_kernel(const float* __restrict__ gi, const float* __restrict__ gh,
                           const float* __restrict__ prev, float* __restrict__ out,
                           int Nn, int H)
{
    int t = blockIdx.x * blockDim.x + threadIdx.x;
    if (t >= Nn * H) return;
    int i = t / H;
    int j = t - i * H;
    size_t b = (size_t)i * 3 * H + j;
    float ir = gi[b], iz = gi[b + H], in_ = gi[b + 2 * H];
    float hr = gh[b], hz = gh[b + H], hn  = gh[b + 2 * H];
    float r  = 1.f / (1.f + __expf(-(ir + hr)));
    float z  = 1.f / (1.f + __expf(-(iz + hz)));
    float nc = tanhf(in_ + r * hn);
    out[t] = (1.f - z) * nc + z * prev[t];
}

// Row-wise L2 normalize: one wave32 per row, shfl_xor reduction
__global__ void l2norm_kernel(const float* __restrict__ h, float* __restrict__ out,
                              int Nn, int H)
{
    int lane = threadIdx.x & 31;
    int row  = (blockIdx.x * blockDim.x + threadIdx.x) >> 5;
    if (row >= Nn) return;
    const float* hp = h + (size_t)row * H;
    float s = 0.f;
    for (int j = lane; j < H; j += 32) { float v = hp[j]; s += v * v; }
#pragma unroll
    for (int off = 16; off > 0; off >>= 1) s += __shfl_xor(s, off, 32);
    float rn = rsqrtf(s);
    float* op = out + (size_t)row * H;
    for (int j = lane; j < H; j += 32) op[j] = hp[j] * rn;
}

// ---------------------------------------------------------------------------
// Host side
// ---------------------------------------------------------------------------
static inline void launch_gemm(const float* A, const float* W, const float* bias,
                               float* out, int M, int K, int Ncols,
                               bool act, bool hasBias, hipStream_t stream)
{
    const int ntiles = Ncols / 16;
    const int NT = (ntiles % 4 == 0) ? 4 : 2;   // all Ncols here are /32
    const int totGrps = (M / 16) * (ntiles / NT);
    dim3 grid((totGrps + 7) / 8), block(256);
    if (NT == 4) {
        if (act)          wmma_gemm_f32_kernel<1, 1, 4><<<grid, block, 0, stream>>>(A, W, bias, out, M, K, Ncols);
        else if (hasBias) wmma_gemm_f32_kernel<0, 1, 4><<<grid, block, 0, stream>>>(A, W, bias, out, M, K, Ncols);
        else              wmma_gemm_f32_kernel<0, 0, 4><<<grid, block, 0, stream>>>(A, W, bias, out, M, K, Ncols);
    } else {
        if (act)          wmma_gemm_f32_kernel<1, 1, 2><<<grid, block, 0, stream>>>(A, W, bias, out, M, K, Ncols);
        else if (hasBias) wmma_gemm_f32_kernel<0, 1, 2><<<grid, block, 0, stream>>>(A, W, bias, out, M, K, Ncols);
        else              wmma_gemm_f32_kernel<0, 0, 2><<<grid, block, 0, stream>>>(A, W, bias, out, M, K, Ncols);
    }
}

static inline int cdiv(long long a, int b) { return (int)((a + b - 1) / b); }

extern "C" void kernel_launch(void* const* d_in, const int* in_sizes, int n_in,
                              void* d_out, int out_size, void* d_ws, size_t ws_size,
                              hipStream_t stream)
{
    const float* x     = (const float*)d_in[0];
    const float* prev0 = (const float*)d_in[1];
    const float* prev1 = (const float*)d_in[2];
    const float* W_p1  = (const float*)d_in[3];
    const float* b_p1  = (const float*)d_in[4];
    const float* W_p2  = (const float*)d_in[5];
    const float* b_p2  = (const float*)d_in[6];
    const float* W_c1  = (const float*)d_in[7];
    const float* b_c1  = (const float*)d_in[8];
    const float* W_c2  = (const float*)d_in[9];
    const float* b_c2  = (const float*)d_in[10];
    const float* Wih1  = (const float*)d_in[11];
    const float* Whh1  = (const float*)d_in[12];
    const float* bih1  = (const float*)d_in[13];
    const float* bhh1  = (const float*)d_in[14];
    const float* Wih2  = (const float*)d_in[15];
    const float* Whh2  = (const float*)d_in[16];
    const float* bih2  = (const float*)d_in[17];
    const float* bhh2  = (const float*)d_in[18];
    const int*   srcI  = (const int*)d_in[19];
    const int*   dstI  = (const int*)d_in[20];

    const int INF = 256, H1 = 64, H2 = 32;
    const int Nn = in_sizes[0] / INF;       // 100000
    const int E  = in_sizes[19];            // 1600000

    // Workspace layout (floats). Total = Nn*642 floats (~257 MB).
    float* ws   = (float*)d_ws;
    float* X1   = ws;                         // Nn*256 : h1, later gi1/gi2
    float* X2   = X1 + (size_t)Nn * 256;      // Nn*192 : h2, later gh1/gh2
    float* X3   = X2 + (size_t)Nn * 192;      // Nn*64  : hw1, later gru1 out
    float* X4   = X3 + (size_t)Nn * 64;       // Nn*64  : agg1 -> gconv1 (in-place)
    float* degb = X4 + (size_t)Nn * 64;       // Nn
    float* dinv = degb + (size_t)Nn;          // Nn
    float* X7   = dinv + (size_t)Nn;          // Nn*32  : hw2
    float* X8   = X7 + (size_t)Nn * 32;       // Nn*32  : agg2 -> gconv2 (in-place)

    float* hOut = (float*)d_out;              // [Nn,32] final h
    float* emb0 = hOut + (size_t)Nn * H2;     // [Nn,64] normalized emb

    const int TB = 256;

    // ---- preprocess MLP ----
    launch_gemm(x,  W_p1, b_p1, X1, Nn, 256, 256, true, true, stream);   // h1 = leaky(x@Wp1+b)
    launch_gemm(X1, W_p2, b_p2, X2, Nn, 256, 128, true, true, stream);   // h2 = leaky(h1@Wp2+b)

    // ---- degree / D^-1/2 (shared by both convs) ----
    fill_kernel<<<cdiv(Nn, TB), TB, 0, stream>>>(degb, 1.0f, Nn);        // self-loop
    deg_kernel<<<cdiv(E, TB), TB, 0, stream>>>(dstI, degb, E);
    dinv_kernel<<<cdiv(Nn, TB), TB, 0, stream>>>(degb, dinv, Nn);

    // ---- GCNConv(128 -> 64) ----
    launch_gemm(X2, W_c1, nullptr, X3, Nn, 128, H1, false, false, stream);   // hw1
    fill_kernel<<<cdiv((long long)Nn * H1, TB), TB, 0, stream>>>(X4, 0.f, Nn * H1);
    scatter_kernel<64><<<cdiv((long long)E * (H1 / 4), TB), TB, 0, stream>>>(
        X3, srcI, dstI, dinv, X4, E);
    gcn_finalize_kernel<<<cdiv((long long)Nn * H1, TB), TB, 0, stream>>>(
        X4, X3, dinv, b_c1, Nn, H1);                                     // leaky(conv1)

    // ---- GRUCell(64) ----
    launch_gemm(X4,    Wih1, bih1, X1, Nn, H1, 3 * H1, false, true, stream); // gi1
    launch_gemm(prev0, Whh1, bhh1, X2, Nn, H1, 3 * H1, false, true, stream); // gh1
    gru_kernel<<<cdiv((long long)Nn * H1, TB), TB, 0, stream>>>(X1, X2, prev0, X3, Nn, H1);

    // ---- L2 normalize -> emb0 (output 2) ----
    l2norm_kernel<<<cdiv((long long)Nn * 32, TB), TB, 0, stream>>>(X3, emb0, Nn, H1);

    // ---- GCNConv(64 -> 32) ----
    launch_gemm(emb0, W_c2, nullptr, X7, Nn, H1, H2, false, false, stream);  // hw2
    fill_kernel<<<cdiv((long long)Nn * H2, TB), TB, 0, stream>>>(X8, 0.f, Nn * H2);
    scatter_kernel<32><<<cdiv((long long)E * (H2 / 4), TB), TB, 0, stream>>>(
        X7, srcI, dstI, dinv, X8, E);
    gcn_finalize_kernel<<<cdiv((long long)Nn * H2, TB), TB, 0, stream>>>(
        X8, X7, dinv, b_c2, Nn, H2);                                     // leaky(conv2)

    // ---- GRUCell(32) -> final h (output 1) ----
    launch_gemm(X8,    Wih2, bih2, X1, Nn, H2, 3 * H2, false, true, stream); // gi2
    launch_gemm(prev1, Whh2, bhh2, X2, Nn, H2, 3 * H2, false, true, stream); // gh2
    gru_kernel<<<cdiv((long long)Nn * H2, TB), TB, 0, stream>>>(X1, X2, prev1, hOut, Nn, H2);
}